// IAGNN_10634339025457
// MI455X (gfx1250) — compile-verified
//
#include <hip/hip_runtime.h>
#include <hip/hip_bf16.h>
#include <math.h>

#define BDIM 512
#define LDIM 50
#define EDIM 128
#define NIDIM 10
#define NNDIM 50000
#define TOPK 512   // int(0.1 * 512 * 10)

typedef __attribute__((ext_vector_type(2))) float v2f;
typedef __attribute__((ext_vector_type(8))) float v8f;

// ---------------- K0: t1 = w_p @ l1_w^T + l1_b  [NI, E] ----------------
__global__ void k0_t1(const float* __restrict__ w_p, const float* __restrict__ l1w,
                      const float* __restrict__ l1b, float* __restrict__ t1) {
    int t = blockIdx.x * blockDim.x + threadIdx.x;
    if (t >= NIDIM * EDIM) return;
    int i = t / EDIM, j = t % EDIM;
    const float* wr = w_p + i * EDIM;
    const float* lr = l1w + j * EDIM;
    float acc = l1b[j];
#pragma unroll 8
    for (int k = 0; k < EDIM; ++k) acc += wr[k] * lr[k];
    t1[t] = acc;
}

// ---------------- K1: gather + 3 rounds of graph propagation ----------------
// sign(cosine_sim) == sign(raw dot) since inverse norms are strictly positive,
// so H = (dot(cur_l, cur_m) > 0) with no normalization needed.
__global__ void k1_graph(const int* __restrict__ data, const float* __restrict__ emb,
                         float* __restrict__ vout) {
    __shared__ float cur[LDIM * EDIM];
    __shared__ float nxt[LDIM * EDIM];
    __shared__ unsigned Hlo[LDIM], Hhi[LDIM];
    int b = blockIdx.x;
    int tid = threadIdx.x;
    float* vb = vout + (size_t)b * LDIM * EDIM;

    for (int q = tid; q < LDIM * EDIM; q += blockDim.x) {
        int l = q >> 7, e = q & 127;
        int idx = data[b * LDIM + l];
        float val = (idx > 0) ? emb[(size_t)(idx - 1) * EDIM + e] : 0.0f;
        cur[q] = val;
        vb[q] = val;
    }
    for (int it = 0; it < 3; ++it) {
        if (tid < LDIM) { Hlo[tid] = 0u; Hhi[tid] = 0u; }
        __syncthreads();
        // adjacency bitmasks via pairwise dot signs
        for (int p = tid; p < LDIM * LDIM; p += blockDim.x) {
            int l = p / LDIM, m = p % LDIM;
            const float* rl = &cur[l * EDIM];
            const float* rm = &cur[m * EDIM];
            float d = 0.f;
#pragma unroll 8
            for (int k = 0; k < EDIM; ++k) d += rl[k] * rm[k];
            if (d > 0.f) {
                if (m < 32) atomicOr(&Hlo[l], 1u << m);
                else        atomicOr(&Hhi[l], 1u << (m - 32));
            }
        }
        __syncthreads();
        // nxt = D^-1 H cur
        for (int q = tid; q < LDIM * EDIM; q += blockDim.x) {
            int l = q >> 7, e = q & 127;
            unsigned lo = Hlo[l], hi = Hhi[l];
            int deg = __popc(lo) + __popc(hi);
            float acc = 0.f;
            while (lo) { int m = __ffs(lo) - 1; lo &= lo - 1; acc += cur[m * EDIM + e]; }
            while (hi) { int m = __ffs(hi) - 1; hi &= hi - 1; acc += cur[(m + 32) * EDIM + e]; }
            nxt[q] = (deg > 0) ? acc / (float)deg : 0.f;
        }
        __syncthreads();
        for (int q = tid; q < LDIM * EDIM; q += blockDim.x) {
            vb[q] += nxt[q];
            cur[q] = nxt[q];
        }
        __syncthreads();
    }
}

// ---------------- block reduce over 128 threads (4 x wave32) ----------------
__device__ __forceinline__ float blockReduce128(float v, volatile float* red) {
#pragma unroll
    for (int o = 16; o > 0; o >>= 1) v += __shfl_down(v, o, 32);
    __syncthreads();
    if ((threadIdx.x & 31) == 0) red[threadIdx.x >> 5] = v;
    __syncthreads();
    return red[0] + red[1] + red[2] + red[3];
}

// ---------------- K2: intention attention: v -> C -> m -> alpha -> beta ----------------
__global__ void k2_intent(const int* __restrict__ data, const float* __restrict__ vws,
                          const float* __restrict__ t1g, const float* __restrict__ w1,
                          const float* __restrict__ qalpha,
                          const float* __restrict__ l2w, const float* __restrict__ l2b,
                          const float* __restrict__ l3w, const float* __restrict__ l3b,
                          float* __restrict__ mg, float* __restrict__ betag) {
    __shared__ float vb[LDIM * EDIM];
    __shared__ float t1s[NIDIM * EDIM];
    __shared__ float Cm[NIDIM * LDIM];
    __shared__ float msk[LDIM];
    __shared__ float csum[NIDIM];
    __shared__ float mi[EDIM];
    __shared__ float red[4];
    __shared__ float al[NIDIM];
    int b = blockIdx.x;
    int e = threadIdx.x;  // 128 threads
    const float* vsrc = vws + (size_t)b * LDIM * EDIM;
    for (int q = e; q < LDIM * EDIM; q += 128) vb[q] = vsrc[q];
    for (int q = e; q < NIDIM * EDIM; q += 128) t1s[q] = t1g[q];
    if (e < LDIM) msk[e] = (data[b * LDIM + e] != 0) ? 1.f : 0.f;
    __syncthreads();

    float w1e = w1[e], qae = qalpha[e], l2be = l2b[e];
    const float* l2r = l2w + e * EDIM;
    for (int l = 0; l < LDIM; ++l) {
        const float* vrow = &vb[l * EDIM];
        float t2e = l2be;
#pragma unroll 8
        for (int k = 0; k < EDIM; ++k) t2e += l2r[k] * vrow[k];
        for (int i = 0; i < NIDIM; ++i) {
            float u = t1s[i * EDIM + e] + t2e;
            float contrib = (1.f / (1.f + expf(-u))) * w1e;
            float sum = blockReduce128(contrib, red);
            if (e == 0) Cm[i * LDIM + l] = sum * msk[l];
        }
    }
    __syncthreads();
    if (e < NIDIM) {
        float s = 0.f;
        for (int l = 0; l < LDIM; ++l) s += Cm[e * LDIM + l];
        csum[e] = s;
    }
    __syncthreads();
    for (int q = e; q < NIDIM * LDIM; q += 128) Cm[q] /= csum[q / LDIM];
    __syncthreads();

    float macc[NIDIM];
    for (int i = 0; i < NIDIM; ++i) {
        float acc = 0.f;
        for (int l = 0; l < LDIM; ++l) acc += Cm[i * LDIM + l] * vb[l * EDIM + e];
        macc[i] = acc;
        mg[((size_t)b * NIDIM + i) * EDIM + e] = acc;
    }

    const float* l3r = l3w + e * EDIM;
    float l3be = l3b[e];
    for (int i = 0; i < NIDIM; ++i) {
        __syncthreads();
        mi[e] = macc[i];
        __syncthreads();
        float z = l3be;
#pragma unroll 8
        for (int k = 0; k < EDIM; ++k) z += l3r[k] * mi[k];
        float val = (1.f / (1.f + expf(-z))) * qae;
        float s = blockReduce128(val, red);
        if (e == 0) al[i] = s;
    }
    __syncthreads();
    if (e == 0) {
        float mx = al[0];
        for (int i = 1; i < NIDIM; ++i) mx = fmaxf(mx, al[i]);
        float Z = 0.f, ex[NIDIM];
        for (int i = 0; i < NIDIM; ++i) { ex[i] = expf(al[i] - mx); Z += ex[i]; }
        for (int i = 0; i < NIDIM; ++i) betag[b * NIDIM + i] = ex[i] / Z;
    }
}

// ---------------- K3: exact rank-512 threshold + gamma ----------------
__global__ void k3_eps(const float* __restrict__ betag, float* __restrict__ gammag) {
    __shared__ float bet[BDIM * NIDIM];
    __shared__ float epsS;
    int tid = threadIdx.x;
    for (int q = tid; q < BDIM * NIDIM; q += blockDim.x) bet[q] = betag[q];
    __syncthreads();
    for (int q = tid; q < BDIM * NIDIM; q += blockDim.x) {
        float x = bet[q];
        int g = 0, ge = 0;
        for (int j = 0; j < BDIM * NIDIM; ++j) {
            float y = bet[j];
            g += (y > x);
            ge += (y >= x);
        }
        if (g <= TOPK && ge > TOPK) epsS = x;  // unique value; racers write same x
    }
    __syncthreads();
    float eps = epsS;
    for (int q = tid; q < BDIM * NIDIM; q += blockDim.x)
        gammag[q] = (bet[q] >= eps) ? bet[q] : 0.f;
}

// ---------------- K4: s[b,e] = sum_i gamma[b,i] * m[b,i,e] ----------------
__global__ void k4_s(const float* __restrict__ gammag, const float* __restrict__ mg,
                     float* __restrict__ sg) {
    int b = blockIdx.x, e = threadIdx.x;
    float acc = 0.f;
#pragma unroll
    for (int i = 0; i < NIDIM; ++i)
        acc += gammag[b * NIDIM + i] * mg[((size_t)b * NIDIM + i) * EDIM + e];
    sg[b * EDIM + e] = acc;
}

// ---------------- K5: out = s @ emb^T via V_WMMA_F32_16X16X4_F32 ----------------
#define TILE_N 64
#define SA_LD 132   // padded LDS stride (floats) to break bank conflicts
#define SB_LD 132
__global__ void k5_out_wmma(const float* __restrict__ sg, const float* __restrict__ emb,
                            float* __restrict__ out) {
    __shared__ float sA[16 * SA_LD];
    __shared__ float sB[TILE_N * SB_LD];
    int tid = threadIdx.x;  // 128 threads = 4 wave32
    int mbase = blockIdx.y * 16;
    int nbase = blockIdx.x * TILE_N;

    // stage s tile (16 x 128)
    for (int idx = tid; idx < 16 * 32; idx += 128) {
        int r = idx >> 5, c4 = idx & 31;
        float4 v = *(const float4*)(sg + (size_t)(mbase + r) * EDIM + c4 * 4);
        *(float4*)(&sA[r * SA_LD + c4 * 4]) = v;
    }
    // stage emb tile (64 x 128), clamp OOB rows (store masked later)
    for (int idx = tid; idx < TILE_N * 32; idx += 128) {
        int r = idx >> 5, c4 = idx & 31;
        int grow = nbase + r;
        if (grow >= NNDIM) grow = NNDIM - 1;
        float4 v = *(const float4*)(emb + (size_t)grow * EDIM + c4 * 4);
        *(float4*)(&sB[r * SB_LD + c4 * 4]) = v;
    }
    __syncthreads();

    int lane = tid & 31;
    int wave = tid >> 5;               // each wave owns one 16x16 N sub-tile
    int mrow = lane & 15;
    int khalf = (lane >> 4) * 2;       // lanes 0-15: K+0/1, lanes 16-31: K+2/3
    int nloc = wave * 16 + (lane & 15);

    v8f c = {0.f, 0.f, 0.f, 0.f, 0.f, 0.f, 0.f, 0.f};
#pragma unroll
    for (int k0 = 0; k0 < EDIM; k0 += 4) {
        int kk = k0 + khalf;
        v2f a, bb;
        a.x = sA[mrow * SA_LD + kk];
        a.y = sA[mrow * SA_LD + kk + 1];
        bb.x = sB[nloc * SB_LD + kk];
        bb.y = sB[nloc * SB_LD + kk + 1];
        c = __builtin_amdgcn_wmma_f32_16x16x4_f32(false, a, false, bb, (short)0, c,
                                                  false, false);
    }
    int col = nbase + nloc;
    if (col < NNDIM) {
        int mo = mbase + 8 * (lane >> 4);  // C/D layout: VGPR r -> M=r (lo lanes), M=r+8 (hi)
#pragma unroll
        for (int r = 0; r < 8; ++r)
            out[(size_t)(mo + r) * NNDIM + col] = c[r];
    }
}

extern "C" void kernel_launch(void* const* d_in, const int* in_sizes, int n_in,
                              void* d_out, int out_size, void* d_ws, size_t ws_size,
                              hipStream_t stream) {
    (void)in_sizes; (void)n_in; (void)out_size; (void)ws_size;
    const int*   data   = (const int*)  d_in[0];
    const float* emb    = (const float*)d_in[1];
    const float* w_p    = (const float*)d_in[2];
    const float* w1     = (const float*)d_in[3];
    const float* qalpha = (const float*)d_in[4];
    const float* l1w    = (const float*)d_in[5];
    const float* l1b    = (const float*)d_in[6];
    const float* l2w    = (const float*)d_in[7];
    const float* l2b    = (const float*)d_in[8];
    const float* l3w    = (const float*)d_in[9];
    const float* l3b    = (const float*)d_in[10];
    float* out = (float*)d_out;

    float* ws    = (float*)d_ws;
    float* v     = ws;                                   // B*L*E
    float* t1    = v + (size_t)BDIM * LDIM * EDIM;       // NI*E
    float* m     = t1 + NIDIM * EDIM;                    // B*NI*E
    float* beta  = m + (size_t)BDIM * NIDIM * EDIM;      // B*NI
    float* gamma = beta + BDIM * NIDIM;                  // B*NI
    float* sgbuf = gamma + BDIM * NIDIM;                 // B*E

    k0_t1<<<dim3((NIDIM * EDIM + 255) / 256), dim3(256), 0, stream>>>(w_p, l1w, l1b, t1);
    k1_graph<<<dim3(BDIM), dim3(256), 0, stream>>>(data, emb, v);
    k2_intent<<<dim3(BDIM), dim3(128), 0, stream>>>(data, v, t1, w1, qalpha,
                                                    l2w, l2b, l3w, l3b, m, beta);
    k3_eps<<<dim3(1), dim3(1024), 0, stream>>>(beta, gamma);
    k4_s<<<dim3(BDIM), dim3(EDIM), 0, stream>>>(gamma, m, sgbuf);
    k5_out_wmma<<<dim3((NNDIM + TILE_N - 1) / TILE_N, BDIM / 16), dim3(128), 0, stream>>>(
        sgbuf, emb, out);
}